// HeteroDotProductPredictor_71743133712746
// MI455X (gfx1250) — compile-verified
//
#include <hip/hip_runtime.h>

typedef __attribute__((ext_vector_type(2))) float v2f;
typedef __attribute__((ext_vector_type(8))) float v8f;

#define D_FEAT 128

// One wave32 computes 16 edge dot-products via V_WMMA_F32_16X16X4_F32.
// A = 16 gathered src rows (16x4 f32 chunk), B = 16 gathered dst rows
// (4x16 f32 chunk, same lane addressing by symmetry of the ISA layouts).
// Accumulate over K=128 in steps of 4; edge scores are the diagonal of C.
__global__ __launch_bounds__(256) void edge_dot_wmma_kernel(
    const float* __restrict__ h,
    const int*   __restrict__ src,
    const int*   __restrict__ dst,
    float*       __restrict__ out,
    int n_edges)
{
    const int lane = threadIdx.x & 31;
    const int wave = threadIdx.x >> 5;
    const long long group = (long long)blockIdx.x * (blockDim.x >> 5) + wave;
    const long long base  = group * 16;     // first edge of this wave's group
    if (base >= (long long)n_edges) return; // wave-uniform exit, EXEC stays full

    const int m  = lane & 15;   // edge-in-group (A row M / B col N)
    const int hi = lane >> 4;   // half-wave selects K pair {0,1} vs {2,3}

    long long e = base + m;
    if (e >= (long long)n_edges) e = (long long)n_edges - 1; // clamp, no divergence

    const long long srow = (long long)src[e] * D_FEAT;
    const long long drow = (long long)dst[e] * D_FEAT;
    const float* ps = h + srow + 2 * hi;
    const float* pd = h + drow + 2 * hi;

    v8f c = {0.f, 0.f, 0.f, 0.f, 0.f, 0.f, 0.f, 0.f};

#pragma unroll 8
    for (int k0 = 0; k0 < D_FEAT; k0 += 4) {
        // A vgpr j <- ps[k0 + 2*hi + j], B vgpr j <- pd[k0 + 2*hi + j]:
        // both are one contiguous 8B load per lane (global_load_b64).
        v2f a = *(const v2f*)(ps + k0);
        v2f b = *(const v2f*)(pd + k0);
        // 8 args: (neg_a, A, neg_b, B, c_mod, C, reuse_a, reuse_b)
        c = __builtin_amdgcn_wmma_f32_16x16x4_f32(
                false, a, false, b, (short)0, c, false, false);
    }

    // Diagonal extraction per documented 16x16 f32 C/D layout:
    //   lanes 0..15  : N = lane,    vgpr r holds M = r      -> diag at lanes 0..7,  c[lane]
    //   lanes 16..31 : N = lane-16, vgpr r holds M = 8 + r  -> diag at lanes 24..31, c[lane-24]
    const bool lo  = (lane < 8);
    const bool hid = (lane >= 24);
    const int  v   = (lo ? lane : (lane - 24)) & 7;
    float val;
    switch (v) {
        case 0: val = c[0]; break;
        case 1: val = c[1]; break;
        case 2: val = c[2]; break;
        case 3: val = c[3]; break;
        case 4: val = c[4]; break;
        case 5: val = c[5]; break;
        case 6: val = c[6]; break;
        default: val = c[7]; break;
    }
    const long long eo = base + (lo ? lane : (lane - 16));
    if ((lo || hid) && eo < (long long)n_edges) out[eo] = val;
}

extern "C" void kernel_launch(void* const* d_in, const int* in_sizes, int n_in,
                              void* d_out, int out_size, void* d_ws, size_t ws_size,
                              hipStream_t stream) {
    const float* h   = (const float*)d_in[0];
    const int*   src = (const int*)d_in[1];
    const int*   dst = (const int*)d_in[2];
    float*       out = (float*)d_out;
    const int n_edges = in_sizes[1];

    const long long groups = ((long long)n_edges + 15) / 16; // 16 edges per wave
    const int waves_per_block = 8;                           // 256 threads, wave32
    int blocks = (int)((groups + waves_per_block - 1) / waves_per_block);
    if (blocks < 1) blocks = 1;

    edge_dot_wmma_kernel<<<blocks, 256, 0, stream>>>(h, src, dst, out, n_edges);
}